// TPMessage_50122268344443
// MI455X (gfx1250) — compile-verified
//
#include <hip/hip_runtime.h>
#include <math.h>

// ---------------------------------------------------------------------------
// Equivariant GNN message-passing layer for MI455X (gfx1250, wave32).
// Dominant cost: per-edge fully-connected tensor product weight contraction
// (31.7 GFLOP) -> V_WMMA_F32_16X16X4_F32 (exact fp32 path):
//   M = 16 edges per wave tile, K = u (4/step), N = 16 output channels.
// Edge rows staged via GLOBAL_LOAD_ASYNC_TO_LDS_B128 (ASYNCcnt) and gated
// in place in LDS; tmp (WMMA A operand) staged in LDS in native A layout.
// Working set is L2-resident (38MB << 192MB L2) -> compute/L2-bound.
// ---------------------------------------------------------------------------

typedef __attribute__((ext_vector_type(2))) float v2f;
typedef __attribute__((ext_vector_type(8))) float v8f;

#define N_NODES   10000
#define N_EDGES   100000
#define NODE_DIM  128
#define NUM_BASIS 20
#define NUM_IRREPS 224
#define HIDDEN    352
#define SPH_DIM   480
#define LN_EPS    1e-5f

#define SROW 484   // padded LDS row stride (floats): 16B-aligned, conflict-free

// ws layout (floats)
#define WS_SPH_OFF   0          // spherical_in  : 10000*480
#define WS_SOUT_OFF  4800000    // scalar_out    : 10000*352
#define WS_CG_OFF    8320000    // CG tables     : 615
#define WS_TPW_OFF   8320640    // pair-interleaved tp_w : 64512
#define TPW_TOTAL    64512

__device__ __forceinline__ void atomic_add_f32(float* p, float v) {
  (void)__hip_atomic_fetch_add(p, v, __ATOMIC_RELAXED, __HIP_MEMORY_SCOPE_AGENT);
}

__device__ __forceinline__ v8f zero8() {
  v8f z = {0.f,0.f,0.f,0.f,0.f,0.f,0.f,0.f};
  return z;
}

// ---------------------------------------------------------------------------
// Kernel 0a: build normalized Clebsch-Gordan tables on device (615 floats).
// ---------------------------------------------------------------------------
__global__ void cg_init_kernel(float* __restrict__ cg) {
  if (threadIdx.x != 0 || blockIdx.x != 0) return;
  float S[5][3][3] = {};
  const float r2 = 0.70710678118654752f;
  const float s6 = 0.40824829046386302f;  // 1/sqrt(6)
  S[0][0][1] = r2;  S[0][1][0] = r2;
  S[1][1][2] = r2;  S[1][2][1] = r2;
  S[2][0][0] = -s6; S[2][1][1] = -s6; S[2][2][2] = 2.0f*s6;
  S[3][0][2] = r2;  S[3][2][0] = r2;
  S[4][0][0] = r2;  S[4][1][1] = -r2;
  float E3[3][3][3] = {};
  E3[0][1][2] = 1.f;  E3[1][2][0] = 1.f;  E3[2][0][1] = 1.f;
  E3[0][2][1] = -1.f; E3[2][1][0] = -1.f; E3[1][0][2] = -1.f;
  float t[125];
  auto emit = [&](int off, int n) {
    float ss = 0.f;
    for (int i = 0; i < n; i++) ss += t[i]*t[i];
    float inv = rsqrtf(ss);
    for (int i = 0; i < n; i++) cg[off+i] = t[i]*inv;
  };
  t[0] = 1.f; emit(0, 1);                                   // (0,0,0) @0
  for (int m=0;m<3;m++) for (int n=0;n<3;n++) t[m*3+n] = (m==n)?1.f:0.f;
  emit(1, 9);                                               // (1,1,0) @1
  for (int m=0;m<5;m++) for (int n=0;n<5;n++) t[m*5+n] = (m==n)?1.f:0.f;
  emit(10, 25);                                             // (2,2,0) @10
  for (int n=0;n<3;n++) for (int k=0;k<3;k++) t[n*3+k] = (n==k)?1.f:0.f;
  emit(35, 9);                                              // (0,1,1) @35
  for (int m=0;m<3;m++) for (int k=0;k<3;k++) t[m*3+k] = (m==k)?1.f:0.f;
  emit(44, 9);                                              // (1,0,1) @44
  for (int m=0;m<3;m++) for (int n=0;n<3;n++) for (int k=0;k<3;k++)
    t[(m*3+n)*3+k] = E3[m][n][k];
  emit(53, 27);                                             // (1,1,1) @53
  for (int m=0;m<3;m++) for (int n=0;n<5;n++) for (int k=0;k<3;k++)
    t[(m*5+n)*3+k] = S[n][m][k];
  emit(80, 45);                                             // (1,2,1) @80
  for (int M=0;M<5;M++) for (int n=0;n<3;n++) for (int k=0;k<3;k++)
    t[(M*3+n)*3+k] = S[M][n][k];
  emit(125, 45);                                            // (2,1,1) @125
  for (int M=0;M<5;M++) for (int N=0;N<5;N++) for (int k=0;k<3;k++) {
    float a = 0.f;
    for (int i=0;i<3;i++) for (int j=0;j<3;j++) for (int l=0;l<3;l++)
      a += E3[i][j][k]*S[M][i][l]*S[N][l][j];
    t[(M*5+N)*3+k] = a;
  }
  emit(170, 75);                                            // (2,2,1) @170
  for (int n=0;n<5;n++) for (int k=0;k<5;k++) t[n*5+k] = (n==k)?1.f:0.f;
  emit(245, 25);                                            // (0,2,2) @245
  for (int m=0;m<3;m++) for (int n=0;n<3;n++) for (int k=0;k<5;k++)
    t[(m*3+n)*5+k] = S[k][m][n];
  emit(270, 45);                                            // (1,1,2) @270
  for (int i=0;i<3;i++) for (int M=0;M<5;M++) for (int N=0;N<5;N++) {
    float a = 0.f;
    for (int p=0;p<3;p++) for (int q=0;q<3;q++) for (int c=0;c<3;c++)
      a += E3[i][p][q]*S[M][p][c]*S[N][c][q];
    t[(i*5+M)*5+N] = a;
  }
  emit(315, 75);                                            // (1,2,2) @315
  for (int m=0;m<5;m++) for (int k=0;k<5;k++) t[m*5+k] = (m==k)?1.f:0.f;
  emit(390, 25);                                            // (2,0,2) @390
  for (int M=0;M<5;M++) for (int j=0;j<3;j++) for (int N=0;N<5;N++) {
    float a = 0.f;
    for (int p=0;p<3;p++) for (int q=0;q<3;q++) for (int c=0;c<3;c++)
      a += E3[j][p][q]*S[M][p][c]*S[N][c][q];
    t[(M*3+j)*5+N] = a;
  }
  emit(415, 75);                                            // (2,1,2) @415
  for (int M=0;M<5;M++) for (int N=0;N<5;N++) for (int P=0;P<5;P++) {
    float a = 0.f;
    for (int p=0;p<3;p++) for (int q=0;q<3;q++) for (int c=0;c<3;c++)
      a += S[M][p][q]*S[N][q][c]*S[P][c][p];
    t[(M*5+N)*5+P] = a;
  }
  emit(490, 125);                                           // (2,2,2) @490
}

// ---------------------------------------------------------------------------
// Kernel 0b: pair-interleave tp_w so each WMMA K-step's B operand is one b64:
//   dst[WOFF + ((u>>1)*M3 + w)*2 + (u&1)] = src[WOFF + u*M3 + w]
// ---------------------------------------------------------------------------
__constant__ int PATH_OFF[16] = {0,16384,24576,28672,36864,40960,45056,49152,
                                 51200,53248,57344,59392,61440,62464,63488,64512};
__constant__ int PATH_M3[15]  = {128,128,128,64,64,64,64,64,64,32,32,32,32,32,32};

__global__ __launch_bounds__(256) void tpw_shuffle_kernel(
    const float* __restrict__ src, float* __restrict__ dst)
{
  int idx = blockIdx.x*256 + threadIdx.x;
  if (idx >= TPW_TOTAL) return;
  int p = 0;
  while (idx >= PATH_OFF[p+1]) p++;
  int loc = idx - PATH_OFF[p];
  int m3 = PATH_M3[p];
  int u = loc / m3, w = loc % m3;
  dst[PATH_OFF[p] + ((u>>1)*m3 + w)*2 + (u&1)] = src[idx];
}

// ---------------------------------------------------------------------------
// Block reduction helper (128 threads).
// ---------------------------------------------------------------------------
__device__ __forceinline__ float bsum(float v, float* red) {
  int t = threadIdx.x;
  __syncthreads();
  red[t] = v;
  __syncthreads();
  for (int off = 64; off > 0; off >>= 1) {
    if (t < off) red[t] += red[t + off];
    __syncthreads();
  }
  return red[0];
}

// ---------------------------------------------------------------------------
// Kernel 1: per-node LN + O3-LN + MLP. One 128-thread block per node.
// ---------------------------------------------------------------------------
__global__ __launch_bounds__(128) void node_kernel(
    const float* __restrict__ x_scalar, const float* __restrict__ x_sph,
    const float* __restrict__ ln_w, const float* __restrict__ ln_b,
    const float* __restrict__ o3_w, const float* __restrict__ o3_b,
    const float* __restrict__ w1, const float* __restrict__ b1,
    const float* __restrict__ w2, const float* __restrict__ b2,
    float* __restrict__ ws_sph, float* __restrict__ ws_sout,
    float* __restrict__ out_scalar, float* __restrict__ out_sph)
{
  int node = blockIdx.x, t = threadIdx.x;
  __shared__ float red[128];
  __shared__ float s_sin[128];
  __shared__ float s_h[128];
  __shared__ float s_sph[480];

  float x = x_scalar[(size_t)node*NODE_DIM + t];
  float mu  = bsum(x, red) * (1.f/128.f);
  float d   = x - mu;
  float var = bsum(d*d, red) * (1.f/128.f);
  float si  = d * rsqrtf(var + LN_EPS) * ln_w[t] + ln_b[t];
  s_sin[t] = si;
  out_scalar[(size_t)node*NODE_DIM + t] = si;   // accumulation base

  for (int i = t; i < SPH_DIM; i += 128) s_sph[i] = x_sph[(size_t)node*SPH_DIM + i];
  __syncthreads();

  size_t sb = (size_t)node * SPH_DIM;
  // l = 0
  float f0 = s_sph[t];
  float mu0 = bsum(f0, red) * (1.f/128.f);
  f0 -= mu0;
  float n0 = bsum(f0*f0, red) * (1.f/128.f);
  float g0 = f0 * rsqrtf(n0 + LN_EPS) * o3_w[t] + o3_b[t];
  ws_sph[sb + t] = g0; out_sph[sb + t] = g0;
  // l = 1 (192 comps)
  float a1 = s_sph[128 + t];
  float c1 = (t < 64) ? s_sph[256 + t] : 0.f;
  float n1 = bsum(a1*a1 + c1*c1, red) * (1.f/192.f);
  float r1 = rsqrtf(n1 + LN_EPS);
  float o1a = a1 * r1 * o3_w[128 + t/3];
  ws_sph[sb + 128 + t] = o1a; out_sph[sb + 128 + t] = o1a;
  if (t < 64) {
    float o1b = c1 * r1 * o3_w[128 + (128 + t)/3];
    ws_sph[sb + 256 + t] = o1b; out_sph[sb + 256 + t] = o1b;
  }
  // l = 2 (160 comps)
  float a2 = s_sph[320 + t];
  float c2 = (t < 32) ? s_sph[448 + t] : 0.f;
  float n2 = bsum(a2*a2 + c2*c2, red) * (1.f/160.f);
  float r2v = rsqrtf(n2 + LN_EPS);
  float o2a = a2 * r2v * o3_w[192 + t/5];
  ws_sph[sb + 320 + t] = o2a; out_sph[sb + 320 + t] = o2a;
  if (t < 32) {
    float o2b = c2 * r2v * o3_w[192 + (128 + t)/5];
    ws_sph[sb + 448 + t] = o2b; out_sph[sb + 448 + t] = o2b;
  }
  __syncthreads();

  // MLP: h = silu(si @ W1 + b1); scalar_out = h @ W2 + b2
  float acc = b1[t];
  for (int j = 0; j < 128; j++) acc += s_sin[j] * w1[j*128 + t];
  float h = acc / (1.f + __expf(-acc));
  s_h[t] = h;
  __syncthreads();
  for (int c = t; c < HIDDEN; c += 128) {
    float ov = b2[c];
    for (int j = 0; j < 128; j++) ov += s_h[j] * w2[j*HIDDEN + c];
    ws_sout[(size_t)node*HIDDEN + c] = ov;
  }
}

// ---------------------------------------------------------------------------
// Kernel 2: scalar messages (fo[:,224:352]) scatter-added into d_out scalar.
// ---------------------------------------------------------------------------
__global__ __launch_bounds__(128) void msg_scalar_kernel(
    const float* __restrict__ rbf, const float* __restrict__ rbf_w,
    const float* __restrict__ scalar_out, const int* __restrict__ edge_index,
    float* __restrict__ out_scalar)
{
  int e = blockIdx.x, t = threadIdx.x;
  __shared__ float rb_s[NUM_BASIS];
  if (t < NUM_BASIS) rb_s[t] = rbf[(size_t)e*NUM_BASIS + t];
  __syncthreads();
  int dst = edge_index[e];
  int src = edge_index[N_EDGES + e];
  int c = NUM_IRREPS + t;
  float f = 0.f;
  #pragma unroll
  for (int b = 0; b < NUM_BASIS; b++) f += rb_s[b] * rbf_w[b*HIDDEN + c];
  float v = scalar_out[(size_t)src*HIDDEN + c] * f;
  atomic_add_f32(out_scalar + (size_t)dst*NODE_DIM + t, v);
}

// ---------------------------------------------------------------------------
// Kernel 3: fully-connected tensor product via f32 WMMA.
// One wave32 per 16-edge tile; lane = (el = lane&15, half = lane>>4), where
// half selects the K-pair {0,1}/{2,3} of each WMMA K=4 step (u mod 4).
// ---------------------------------------------------------------------------
template <int L1, int L2, int L3, int M1, int M3, int WOFF, int CGOFF,
          int KLO, int KD>
__device__ __forceinline__ void do_path(
    const float* __restrict__ tpw2,       // pair-interleaved weights
    const float* __restrict__ rs,         // 9 rsh values (registers)
    const float* cg_s, const float* s_row, float* tmp_s,
    float* __restrict__ out_sph, const int* dstv,
    int lane, int el, int half)
{
  constexpr int D1 = 2*L1 + 1, D2 = 2*L2 + 1, D3 = 2*L3 + 1;
  constexpr int NS = M1 / 4;               // K steps (K=4 per WMMA)
  constexpr int SPHOFF = (L1==0) ? 0 : ((L1==1) ? 128 : 320);
  constexpr int RSOFF  = (L2==0) ? 0 : ((L2==1) ? 1 : 4);
  constexpr int OUTOFF = (L3==0) ? 0 : ((L3==1) ? 128 : 320);
  const float FANV = (L3==0) ? 224.f : ((L3==1) ? 384.f : 352.f);
  const float pw = sqrtf((float)D3 / FANV);

  // y2[m][kk] = pw * sum_n rsh[n] * CG[m][n][KLO+kk]
  float y2[D1*KD];
  #pragma unroll
  for (int m = 0; m < D1; m++)
    #pragma unroll
    for (int kk = 0; kk < KD; kk++) {
      float a = 0.f;
      #pragma unroll
      for (int n = 0; n < D2; n++)
        a += rs[RSOFF + n] * cg_s[CGOFF + (m*D2 + n)*D3 + (KLO + kk)];
      y2[m*KD + kk] = a * pw;
    }

  // Phase A: tmp[e][u][kk] = sum_m x1[u][m] * y2[m][kk]  (x1 pre-gated in LDS)
  // Stored in WMMA-A layout: flat(((kk*NS+s)*2+half)*16+el)*2+t.
  const float* rowp = s_row + el*SROW + SPHOFF;
  for (int s = 0; s < NS; s++) {
    int u0 = 4*s + 2*half;
    float xv[2][D1];
    #pragma unroll
    for (int tt = 0; tt < 2; tt++)
      #pragma unroll
      for (int m = 0; m < D1; m++)
        xv[tt][m] = rowp[(u0 + tt)*D1 + m];
    #pragma unroll
    for (int kk = 0; kk < KD; kk++) {
      float t0 = 0.f, t1 = 0.f;
      #pragma unroll
      for (int m = 0; m < D1; m++) {
        t0 += xv[0][m] * y2[m*KD + kk];
        t1 += xv[1][m] * y2[m*KD + kk];
      }
      v2f tv; tv.x = t0; tv.y = t1;
      *(v2f*)&tmp_s[((kk*NS + s)*2 + half)*32 + el*2] = tv;
    }
  }
  __syncthreads();

  // Phase B: D[e][w][kk] += A(tmp) x B(W); B is one b64 per K-step.
  for (int wt = 0; wt < M3/16; wt++) {
    v8f acc[KD];
    #pragma unroll
    for (int kk = 0; kk < KD; kk++) acc[kk] = zero8();
    const float* wp = tpw2 + WOFF + (wt*16 + el)*2;
    for (int s = 0; s < NS; s++) {
      v2f B = *(const v2f*)&wp[(size_t)(2*s + half) * (M3*2)];
      #pragma unroll
      for (int kk = 0; kk < KD; kk++) {
        v2f A = *(const v2f*)&tmp_s[((kk*NS + s)*2 + half)*32 + el*2];
        acc[kk] = __builtin_amdgcn_wmma_f32_16x16x4_f32(
            false, A, false, B, (short)0, acc[kk], false, false);
      }
    }
    // D layout: acc[kk][j] = out[e = j + 8*half][w = wt*16 + el][KLO+kk]
    #pragma unroll
    for (int kk = 0; kk < KD; kk++) {
      int col = OUTOFF + (wt*16 + el)*D3 + (KLO + kk);
      #pragma unroll
      for (int j = 0; j < 8; j++)
        atomic_add_f32(out_sph + (size_t)dstv[j]*SPH_DIM + col, acc[kk][j]);
    }
  }
  __syncthreads();
}

__global__ __launch_bounds__(32) void tp_kernel(
    const float* __restrict__ sph_in, const float* __restrict__ scalar_out,
    const float* __restrict__ cg, const float* __restrict__ rbf,
    const float* __restrict__ rsh, const int* __restrict__ edge_index,
    const float* __restrict__ rbf_w, const float* __restrict__ tpw2,
    float* __restrict__ out_sph)
{
  __shared__ float s_cg[616];                       // CG tables
  __shared__ __align__(16) float s_row[16*SROW];    // gated x1 rows (async-staged)
  __shared__ __align__(16) float s_tmp[6144];       // WMMA-A staging

  int lane = threadIdx.x;
  int tile = blockIdx.x;                            // 6250 tiles of 16 edges
  for (int i = lane; i < 615; i += 32) s_cg[i] = cg[i];

  int el = lane & 15, half = lane >> 4;
  int e = tile*16 + el;
  int src = edge_index[N_EDGES + e];
  int dstv[8];
  #pragma unroll
  for (int j = 0; j < 8; j++) dstv[j] = edge_index[tile*16 + j + 8*half];

  float rb[NUM_BASIS];
  #pragma unroll
  for (int b = 0; b < NUM_BASIS; b++) rb[b] = rbf[(size_t)e*NUM_BASIS + b];
  float rs[9];
  #pragma unroll
  for (int n = 0; n < 9; n++) rs[n] = rsh[(size_t)e*9 + n];

  // --- Async-stage this edge's spherical_in row into LDS (CDNA5 TDM-lite:
  // GLOBAL_LOAD_ASYNC_TO_LDS_B128, tracked with ASYNCcnt). Each half-lane
  // copies half of the 480-float row; offset: applies to both sides.
  {
    unsigned lbase = (unsigned)(uintptr_t)&s_row[el*SROW + half*240];
    unsigned long long ga =
        (unsigned long long)(uintptr_t)(sph_in + (size_t)src*SPH_DIM + half*240);
    #pragma unroll
    for (int i = 0; i < 60; i++)
      asm volatile("global_load_async_to_lds_b128 %0, %1, off offset:%2"
                   :: "v"(lbase), "v"(ga), "i"(i*16) : "memory");
    asm volatile("s_wait_asynccnt 0x0" ::: "memory");
  }

  // --- Gate in place: x1[u][m] = sph[u][m] * (scalar_out[src]*filt)[chan(u)],
  // each lane owns the u's with (u mod 4) in {2*half, 2*half+1}.
  const float* so = scalar_out + (size_t)src*HIDDEN;
  float* rp = &s_row[el*SROW];
  #pragma unroll 1
  for (int i = 0; i < 64; i++) {            // l1 = 0
    int u = 4*(i>>1) + 2*half + (i&1);
    float f = 0.f;
    #pragma unroll
    for (int b = 0; b < NUM_BASIS; b++) f += rb[b]*rbf_w[b*HIDDEN + u];
    f *= so[u];
    rp[u] *= f;
  }
  #pragma unroll 1
  for (int i = 0; i < 32; i++) {            // l1 = 1
    int u = 4*(i>>1) + 2*half + (i&1);
    float f = 0.f;
    #pragma unroll
    for (int b = 0; b < NUM_BASIS; b++) f += rb[b]*rbf_w[b*HIDDEN + 128 + u];
    f *= so[128 + u];
    #pragma unroll
    for (int m = 0; m < 3; m++) rp[128 + u*3 + m] *= f;
  }
  #pragma unroll 1
  for (int i = 0; i < 16; i++) {            // l1 = 2
    int u = 4*(i>>1) + 2*half + (i&1);
    float f = 0.f;
    #pragma unroll
    for (int b = 0; b < NUM_BASIS; b++) f += rb[b]*rbf_w[b*HIDDEN + 192 + u];
    f *= so[192 + u];
    #pragma unroll
    for (int m = 0; m < 5; m++) rp[320 + u*5 + m] *= f;
  }
  __syncthreads();

  // 15 paths; (0,2,2) split in k to bound tmp LDS at 6144 floats.
  do_path<0,0,0,128,128,     0,   0, 0,1>(tpw2, rs, s_cg, s_row, s_tmp, out_sph, dstv, lane, el, half);
  do_path<1,1,0, 64,128, 16384,   1, 0,1>(tpw2, rs, s_cg, s_row, s_tmp, out_sph, dstv, lane, el, half);
  do_path<2,2,0, 32,128, 24576,  10, 0,1>(tpw2, rs, s_cg, s_row, s_tmp, out_sph, dstv, lane, el, half);
  do_path<0,1,1,128, 64, 28672,  35, 0,3>(tpw2, rs, s_cg, s_row, s_tmp, out_sph, dstv, lane, el, half);
  do_path<1,0,1, 64, 64, 36864,  44, 0,3>(tpw2, rs, s_cg, s_row, s_tmp, out_sph, dstv, lane, el, half);
  do_path<1,1,1, 64, 64, 40960,  53, 0,3>(tpw2, rs, s_cg, s_row, s_tmp, out_sph, dstv, lane, el, half);
  do_path<1,2,1, 64, 64, 45056,  80, 0,3>(tpw2, rs, s_cg, s_row, s_tmp, out_sph, dstv, lane, el, half);
  do_path<2,1,1, 32, 64, 49152, 125, 0,3>(tpw2, rs, s_cg, s_row, s_tmp, out_sph, dstv, lane, el, half);
  do_path<2,2,1, 32, 64, 51200, 170, 0,3>(tpw2, rs, s_cg, s_row, s_tmp, out_sph, dstv, lane, el, half);
  do_path<0,2,2,128, 32, 53248, 245, 0,3>(tpw2, rs, s_cg, s_row, s_tmp, out_sph, dstv, lane, el, half);
  do_path<0,2,2,128, 32, 53248, 245, 3,2>(tpw2, rs, s_cg, s_row, s_tmp, out_sph, dstv, lane, el, half);
  do_path<1,1,2, 64, 32, 57344, 270, 0,5>(tpw2, rs, s_cg, s_row, s_tmp, out_sph, dstv, lane, el, half);
  do_path<1,2,2, 64, 32, 59392, 315, 0,5>(tpw2, rs, s_cg, s_row, s_tmp, out_sph, dstv, lane, el, half);
  do_path<2,0,2, 32, 32, 61440, 390, 0,5>(tpw2, rs, s_cg, s_row, s_tmp, out_sph, dstv, lane, el, half);
  do_path<2,1,2, 32, 32, 62464, 415, 0,5>(tpw2, rs, s_cg, s_row, s_tmp, out_sph, dstv, lane, el, half);
  do_path<2,2,2, 32, 32, 63488, 490, 0,5>(tpw2, rs, s_cg, s_row, s_tmp, out_sph, dstv, lane, el, half);
}

// ---------------------------------------------------------------------------
extern "C" void kernel_launch(void* const* d_in, const int* in_sizes, int n_in,
                              void* d_out, int out_size, void* d_ws, size_t ws_size,
                              hipStream_t stream) {
  (void)in_sizes; (void)n_in; (void)out_size; (void)ws_size;
  const float* x_scalar   = (const float*)d_in[0];
  const float* x_spherical= (const float*)d_in[1];
  const float* rbf        = (const float*)d_in[2];
  const float* rsh        = (const float*)d_in[3];
  const int*   edge_index = (const int*)  d_in[4];
  const float* ln_w       = (const float*)d_in[5];
  const float* ln_b       = (const float*)d_in[6];
  const float* o3_w       = (const float*)d_in[7];
  const float* o3_b       = (const float*)d_in[8];
  const float* mlp_w1     = (const float*)d_in[9];
  const float* mlp_b1     = (const float*)d_in[10];
  const float* mlp_w2     = (const float*)d_in[11];
  const float* mlp_b2     = (const float*)d_in[12];
  const float* rbf_w      = (const float*)d_in[13];
  const float* tp_w       = (const float*)d_in[14];

  float* ws       = (float*)d_ws;
  float* ws_sph   = ws + WS_SPH_OFF;    // spherical_in staging
  float* ws_sout  = ws + WS_SOUT_OFF;   // scalar_out staging
  float* ws_cg    = ws + WS_CG_OFF;     // CG tables
  float* ws_tpw   = ws + WS_TPW_OFF;    // pair-interleaved weights

  float* out_scalar = (float*)d_out;                      // 10000 x 128
  float* out_sph    = out_scalar + N_NODES*NODE_DIM;      // 10000 x 480

  cg_init_kernel<<<1, 32, 0, stream>>>(ws_cg);
  tpw_shuffle_kernel<<<(TPW_TOTAL + 255)/256, 256, 0, stream>>>(tp_w, ws_tpw);
  node_kernel<<<N_NODES, 128, 0, stream>>>(
      x_scalar, x_spherical, ln_w, ln_b, o3_w, o3_b,
      mlp_w1, mlp_b1, mlp_w2, mlp_b2,
      ws_sph, ws_sout, out_scalar, out_sph);
  msg_scalar_kernel<<<N_EDGES, 128, 0, stream>>>(
      rbf, rbf_w, ws_sout, edge_index, out_scalar);
  tp_kernel<<<N_EDGES/16, 32, 0, stream>>>(
      ws_sph, ws_sout, ws_cg, rbf, rsh, edge_index, rbf_w, ws_tpw, out_sph);
}